// Downsampler_45311904973439
// MI455X (gfx1250) — compile-verified
//
#include <hip/hip_runtime.h>
#include <cstdint>
#include <cstddef>

// ---------------------------------------------------------------------------
// Adaptive downsample (deformable 3x3, bilinear, reflect pad), MI455X/gfx1250.
//
// Shapes (fixed by reference): img (8,3,1024,1024) f32, coeffs (8,9,256,256) f32
// out (8,3,256,256) f32.  Memory-bound: ~165 MB -> ~7us floor @ 23.3 TB/s.
// Whole image (100 MB) fits in 192 MB L2 -> gathers hit L2; keep streamed-once
// coefficient/output traffic non-temporal, stage coefficients to LDS with the
// gfx1250 async-to-LDS engine (ASYNCcnt) so the gather path owns load-return.
// WMMA is inapplicable: the k-reduction has both operands varying per pixel
// (elementwise multiply-reduce, no shared inner dimension).
// ---------------------------------------------------------------------------

constexpr int B_  = 8;
constexpr int C_  = 3;
constexpr int H_  = 1024;
constexpr int W_  = 1024;
constexpr int HO  = 256;
constexpr int WO  = 256;
constexpr int NK  = 9;            // 3x3 taps
constexpr int HP  = H_ + 2;       // reflect-padded extents (PADDING=1)
constexpr int WP  = W_ + 2;

// Async copy 16B global -> LDS.  ISA 15.18.3 opcode 98 GLOBAL_LOAD_ASYNC_TO_LDS_B128.
// VDST = VGPR holding wave-relative LDS byte address (low 32 bits of a generic
// shared pointer per the aperture mapping), VADDR = 64-bit global address pair.
__device__ __forceinline__ void async_copy_b128(const float* g, float* lds_generic) {
  const unsigned lds_off = (unsigned)(uintptr_t)lds_generic;
  asm volatile("global_load_async_to_lds_b128 %0, %1, off"
               :
               : "v"(lds_off), "v"(g)
               : "memory");
}

__device__ __forceinline__ void wait_async_zero() {
#if __has_builtin(__builtin_amdgcn_s_wait_asynccnt)
  __builtin_amdgcn_s_wait_asynccnt(0);
#else
  asm volatile("s_wait_asynccnt 0" ::: "memory");
#endif
}

// padded index p in [0, n+1] -> original index with reflect (no edge repeat):
// p=0 -> 1 ; p in [1,n] -> p-1 ; p=n+1 -> n-2
__device__ __forceinline__ int reflect_map(int p, int n) {
  int r = p - 1;
  r = (r < 0) ? -r : r;
  r = (r > n - 1) ? (2 * (n - 1) - r) : r;
  return r;
}

__global__ __launch_bounds__(WO)
void adaptive_downsample_kernel(const float* __restrict__ img,
                                const float* __restrict__ kernels,
                                const float* __restrict__ offsets_h,
                                const float* __restrict__ offsets_v,
                                const float* __restrict__ offset_unit,
                                float* __restrict__ out) {
  // LDS: [arr(kern/oh/ov)][k][w]  = 3*9*256 floats = 27 KB
  __shared__ float s_coef[3 * NK * WO];

  const int w = threadIdx.x;     // 0..255 (output column)
  const int h = blockIdx.x;      // 0..255 (output row)
  const int b = blockIdx.y;      // 0..7

  // ---- stage the 27 coefficient rows (1 KB each) into LDS asynchronously ----
  // 27 segments * 64 b128 transfers = 1728 transfers, strided over 256 threads.
  for (int i = threadIdx.x; i < 27 * 64; i += WO) {
    const int seg = i >> 6;            // 0..26
    const int t   = i & 63;            // b128 index within the 1 KB row
    const int arr = seg / NK;          // 0:kernels 1:offsets_h 2:offsets_v
    const int k   = seg - arr * NK;
    const float* src = (arr == 0) ? kernels : (arr == 1) ? offsets_h : offsets_v;
    const float* g = src + (size_t)(((b * NK + k) * HO + h) * WO) + t * 4;
    float*       l = &s_coef[(arr * NK + k) * WO + t * 4];
    async_copy_b128(g, l);
  }
  wait_async_zero();
  __syncthreads();

  const float ou = offset_unit[0];                       // uniform -> s_load
  const float cy = (h + 0.5f) * 4.0f - 0.5f;             // padded-space center
  const float cx = (w + 0.5f) * 4.0f - 0.5f;
  const float* img_b = img + (size_t)b * (C_ * H_ * W_);

  float acc0 = 0.0f, acc1 = 0.0f, acc2 = 0.0f;

#pragma unroll
  for (int k = 0; k < NK; ++k) {
    const int kyi = k / 3;
    const int kxi = k - kyi * 3;

    const float kv = s_coef[(0 * NK + k) * WO + w];
    const float oh = s_coef[(1 * NK + k) * WO + w];
    const float ov = s_coef[(2 * NK + k) * WO + w];

    const float py = cy + (float)kyi + ov * ou;
    const float px = cx + (float)kxi + oh * ou;

    const float y0f = floorf(py);
    const float x0f = floorf(px);
    const float beta  = py - y0f;          // from UNclipped floor (matches ref)
    const float alpha = px - x0f;

    int y0 = (int)y0f; y0 = min(max(y0, 0), HP - 1);
    int y1 = min(y0 + 1, HP - 1);
    int x0 = (int)x0f; x0 = min(max(x0, 0), WP - 1);
    int x1 = min(x0 + 1, WP - 1);

    const int yo0 = reflect_map(y0, H_);
    const int yo1 = reflect_map(y1, H_);
    const int xo0 = reflect_map(x0, W_);
    const int xo1 = reflect_map(x1, W_);

    const float w00 = (1.0f - alpha) * (1.0f - beta);
    const float w01 = alpha * (1.0f - beta);
    const float w10 = (1.0f - alpha) * beta;
    const float w11 = alpha * beta;

    const int i00 = yo0 * W_ + xo0;
    const int i01 = yo0 * W_ + xo1;
    const int i10 = yo1 * W_ + xo0;
    const int i11 = yo1 * W_ + xo1;

    // Gathers: default temporal hint -> image stays resident in 192 MB L2.
    {
      const float* p0 = img_b;
      const float* p1 = img_b + (H_ * W_);
      const float* p2 = img_b + 2 * (H_ * W_);
      const float v0 = w00 * p0[i00] + w01 * p0[i01] + w10 * p0[i10] + w11 * p0[i11];
      const float v1 = w00 * p1[i00] + w01 * p1[i01] + w10 * p1[i10] + w11 * p1[i11];
      const float v2 = w00 * p2[i00] + w01 * p2[i01] + w10 * p2[i10] + w11 * p2[i11];
      acc0 = fmaf(kv, v0, acc0);
      acc1 = fmaf(kv, v1, acc1);
      acc2 = fmaf(kv, v2, acc2);
    }
  }

  // Output written once -> non-temporal, don't pollute L2 image residency.
  const size_t o0 = (((size_t)b * C_ + 0) * HO + h) * WO + w;
  __builtin_nontemporal_store(acc0, &out[o0]);
  __builtin_nontemporal_store(acc1, &out[o0 + (size_t)HO * WO]);
  __builtin_nontemporal_store(acc2, &out[o0 + 2 * (size_t)HO * WO]);
}

extern "C" void kernel_launch(void* const* d_in, const int* in_sizes, int n_in,
                              void* d_out, int out_size, void* d_ws, size_t ws_size,
                              hipStream_t stream) {
  (void)in_sizes; (void)n_in; (void)out_size; (void)d_ws; (void)ws_size;
  const float* img        = (const float*)d_in[0];
  const float* kernels    = (const float*)d_in[1];
  const float* offsets_h  = (const float*)d_in[2];
  const float* offsets_v  = (const float*)d_in[3];
  const float* offset_u   = (const float*)d_in[4];
  float* out              = (float*)d_out;

  dim3 grid(HO, B_);   // one block per (output row, batch)
  dim3 block(WO);      // 256 threads = 8 wave32
  adaptive_downsample_kernel<<<grid, block, 0, stream>>>(
      img, kernels, offsets_h, offsets_v, offset_u, out);
}